// Petri_Cheb_GNN_76639396430230
// MI455X (gfx1250) — compile-verified
//
#include <hip/hip_runtime.h>
#include <hip/hip_bf16.h>

typedef __attribute__((ext_vector_type(2))) float v2f;
typedef __attribute__((ext_vector_type(8))) float v8f;
typedef int v4i_vs __attribute__((vector_size(16)));   // param type of async-LDS builtin

// ---------------- elementwise / scatter kernels ----------------

__global__ void k_degree(const int* __restrict__ src, const float* __restrict__ ew,
                         float* __restrict__ deg, int E) {
    int e = blockIdx.x * blockDim.x + threadIdx.x;
    if (e < E) atomicAdd(&deg[src[e]], ew[e]);
}

__global__ void k_dis(const float* __restrict__ deg, float* __restrict__ dis, int N) {
    int n = blockIdx.x * blockDim.x + threadIdx.x;
    if (n < N) {
        float d = deg[n];
        dis[n] = (d > 0.f) ? rsqrtf(d) : 0.f;
    }
}

__global__ void k_wn(const int* __restrict__ src, const int* __restrict__ tgt,
                     const float* __restrict__ ew, const float* __restrict__ dis,
                     float* __restrict__ wn, int E) {
    int e = blockIdx.x * blockDim.x + threadIdx.x;
    if (e < E) wn[e] = -dis[src[e]] * ew[e] * dis[tgt[e]];
}

// out[tgt[e], :] += wn[e] * H[src[e], :]; one thread handles 2 consecutive features
__global__ void k_prop(const float* __restrict__ H, const int* __restrict__ src,
                       const int* __restrict__ tgt, const float* __restrict__ wn,
                       float* __restrict__ out, int E) {
    long long idx = (long long)blockIdx.x * blockDim.x + threadIdx.x;
    if (idx >= (long long)E * 32) return;
    int e = (int)(idx >> 5);
    int f = ((int)idx & 31) * 2;
    float w = wn[e];
    const float* hp = H + (long long)src[e] * 64 + f;
    float* op = out + (long long)tgt[e] * 64 + f;
    atomicAdd(op,     w * hp[0]);
    atomicAdd(op + 1, w * hp[1]);
}

// P <- 2*P - Tx0   (Chebyshev recurrence, in place on the prop result)
__global__ void k_tx2(const float* __restrict__ Tx0, float* __restrict__ P, int n) {
    int i = blockIdx.x * blockDim.x + threadIdx.x;
    if (i < n) P[i] = 2.f * P[i] - Tx0[i];
}

// ---------------- fused 3-matrix WMMA GEMM ----------------
// Y[N,64] = A0@W[0] + A1@W[1] + A2@W[2] + bias, all fp32, via V_WMMA_F32_16X16X4_F32.
// W[3][64][64] is staged in LDS once per block (async global->LDS when available).
// One wave computes a full 16x64 output strip (4 accumulators); 8 waves / block.
// Requires nrows % 16 == 0.

__device__ __forceinline__ void cheb_accum(const float* __restrict__ A,
                                           const float* __restrict__ sWk,
                                           long long arow, int half, int m,
                                           v8f acc[4])
{
#pragma unroll
    for (int k = 0; k < 64; k += 4) {
        const int ka = k + half * 2;
        // A 16x4: lane m = row, {VGPR0,VGPR1} = K pair, lane-half selects K+2
        v2f a = *reinterpret_cast<const v2f*>(A + arow + ka);
#pragma unroll
        for (int tn = 0; tn < 4; ++tn) {
            // B 4x16: lane m = col, same K striping; W row-major in LDS
            v2f b;
            b.x = sWk[ka * 64 + tn * 16 + m];
            b.y = sWk[(ka + 1) * 64 + tn * 16 + m];
            acc[tn] = __builtin_amdgcn_wmma_f32_16x16x4_f32(
                false, a, false, b, (short)0, acc[tn], false, false);
        }
    }
}

__global__ __launch_bounds__(256) void k_gemm3(
    const float* __restrict__ A0, const float* __restrict__ A1,
    const float* __restrict__ A2,
    const float* __restrict__ Wg,    // [3][64][64]
    const float* __restrict__ bias,  // [64]
    float* __restrict__ Y, int nrows)
{
    __shared__ float sW[3 * 64 * 64];   // 48 KB

    // ---- stage W into LDS (whole block cooperates, before any early-out) ----
#if __has_builtin(__builtin_amdgcn_global_load_async_to_lds_b128) && \
    __has_builtin(__builtin_amdgcn_s_wait_asynccnt)
#pragma unroll
    for (int it = 0; it < 12; ++it) {            // 3072 x 16B chunks / 256 threads
        int t = it * 256 + threadIdx.x;
        __builtin_amdgcn_global_load_async_to_lds_b128(
            (v4i_vs*)(Wg + 4 * t),
            (v4i_vs*)(&sW[4 * t]),
            0, 0);
    }
    __builtin_amdgcn_s_wait_asynccnt(0);
#else
#pragma unroll
    for (int it = 0; it < 12; ++it) {
        int t = it * 256 + threadIdx.x;
        *reinterpret_cast<float4*>(&sW[4 * t]) =
            *reinterpret_cast<const float4*>(Wg + 4 * t);
    }
#endif
    __syncthreads();

    const int wave = threadIdx.x >> 5;
    const int lane = threadIdx.x & 31;
    const int tm   = blockIdx.x * 8 + wave;      // 16-row tile index
    const int rowTiles = nrows >> 4;
    if (tm >= rowTiles) return;                  // wave-uniform: EXEC all-ones for WMMA

    const int m    = lane & 15;
    const int half = lane >> 4;
    const long long arow = (long long)(tm * 16 + m) * 64;

    v8f acc[4] = {};
    cheb_accum(A0, sW,            arow, half, m, acc);
    cheb_accum(A1, sW + 4096,     arow, half, m, acc);
    cheb_accum(A2, sW + 2 * 4096, arow, half, m, acc);

#pragma unroll
    for (int tn = 0; tn < 4; ++tn) {
        const int col = tn * 16 + m;
        const float bv = bias[col];
#pragma unroll
        for (int r = 0; r < 8; ++r) {
            const long long orow = (long long)(tm * 16 + r + half * 8) * 64;
            Y[orow + col] = acc[tn][r] + bv;
        }
    }
}

// ---------------- MLP readout: r = relu(y@wr1+br1)@wr2 + br2 ----------------
__global__ void k_readout(const float* __restrict__ Y, const float* __restrict__ wr1,
                          const float* __restrict__ br1, const float* __restrict__ wr2,
                          const float* __restrict__ br2, float* __restrict__ r, int N)
{
    int n = blockIdx.x * blockDim.x + threadIdx.x;
    if (n >= N) return;
    const float* y = Y + (long long)n * 64;
    float yv[64];
#pragma unroll
    for (int i = 0; i < 64; ++i) yv[i] = y[i];
    float acc = br2[0];
#pragma unroll 1
    for (int j = 0; j < 32; ++j) {
        float s = br1[j];
#pragma unroll
        for (int i = 0; i < 64; ++i) s = fmaf(yv[i], wr1[i * 32 + j], s);
        acc = fmaf(fmaxf(s, 0.f), wr2[j], acc);
    }
    r[n] = acc;
}

// ---------------- scatter-mean pooling ----------------
__global__ void k_pool(const float* __restrict__ r, const int* __restrict__ batch,
                       float* __restrict__ sums, float* __restrict__ cnt, int N) {
    int n = blockIdx.x * blockDim.x + threadIdx.x;
    if (n < N) {
        int g = batch[n];
        atomicAdd(&sums[g], r[n]);
        atomicAdd(&cnt[g], 1.f);
    }
}

__global__ void k_div(const float* __restrict__ sums, const float* __restrict__ cnt,
                      float* __restrict__ out, int G) {
    int g = blockIdx.x * blockDim.x + threadIdx.x;
    if (g < G) out[g] = sums[g] / fmaxf(cnt[g], 1.f);
}

// ---------------- host orchestration ----------------
extern "C" void kernel_launch(void* const* d_in, const int* in_sizes, int n_in,
                              void* d_out, int out_size, void* d_ws, size_t ws_size,
                              hipStream_t stream)
{
    (void)n_in; (void)ws_size;
    const float* x        = (const float*)d_in[0];
    const int*   ei       = (const int*)  d_in[1];
    const float* ew       = (const float*)d_in[2];
    const int*   batch    = (const int*)  d_in[3];
    const float* w_layers = (const float*)d_in[4];
    const float* b_layers = (const float*)d_in[5];
    const float* wr1      = (const float*)d_in[6];
    const float* br1      = (const float*)d_in[7];
    const float* wr2      = (const float*)d_in[8];
    const float* br2      = (const float*)d_in[9];

    const int N = in_sizes[0] / 64;
    const int E = in_sizes[2];
    const int G = out_size;
    const int L = 4;
    const int NF = N * 64;

    const int* src = ei;
    const int* tgt = ei + E;

    float* ws   = (float*)d_ws;
    float* deg  = ws;
    float* dis  = deg + N;
    float* wn   = dis + N;
    float* rbuf = wn + E;
    float* sums = rbuf + N;
    float* cnt  = sums + G;
    float* yA   = cnt + G;
    float* yB   = yA + (size_t)NF;
    float* tx1  = yB + (size_t)NF;
    float* pbf  = tx1 + (size_t)NF;

    dim3 b256(256);
    auto cdiv = [](long long a, long long b) { return (unsigned)((a + b - 1) / b); };

    // Laplacian normalization: deg -> dis -> wn
    (void)hipMemsetAsync(deg, 0, (size_t)N * sizeof(float), stream);
    k_degree<<<cdiv(E, 256), b256, 0, stream>>>(src, ew, deg, E);
    k_dis   <<<cdiv(N, 256), b256, 0, stream>>>(deg, dis, N);
    k_wn    <<<cdiv(E, 256), b256, 0, stream>>>(src, tgt, ew, dis, wn, E);

    const long long propT = (long long)E * 32;
    const float* yPrev = x;
    float* bufs[2] = {yA, yB};
    for (int l = 0; l < L; ++l) {
        (void)hipMemsetAsync(tx1, 0, (size_t)NF * sizeof(float), stream);
        k_prop<<<cdiv(propT, 256), b256, 0, stream>>>(yPrev, src, tgt, wn, tx1, E);
        (void)hipMemsetAsync(pbf, 0, (size_t)NF * sizeof(float), stream);
        k_prop<<<cdiv(propT, 256), b256, 0, stream>>>(tx1, src, tgt, wn, pbf, E);
        k_tx2<<<cdiv(NF, 256), b256, 0, stream>>>(yPrev, pbf, NF);
        float* Ynext = bufs[l & 1];
        int rowTiles = N / 16;
        k_gemm3<<<cdiv(rowTiles, 8), b256, 0, stream>>>(
            yPrev, tx1, pbf,
            w_layers + (size_t)l * 3 * 64 * 64,
            b_layers + (size_t)l * 64,
            Ynext, N);
        yPrev = Ynext;
    }

    k_readout<<<cdiv(N, 256), b256, 0, stream>>>(yPrev, wr1, br1, wr2, br2, rbuf, N);
    (void)hipMemsetAsync(sums, 0, (size_t)G * sizeof(float), stream);
    (void)hipMemsetAsync(cnt,  0, (size_t)G * sizeof(float), stream);
    k_pool<<<cdiv(N, 256), b256, 0, stream>>>(rbuf, batch, sums, cnt, N);
    k_div <<<cdiv(G, 256), b256, 0, stream>>>(sums, cnt, (float*)d_out, G);
}